// MultiBoxLoss_69466801046048
// MI455X (gfx1250) — compile-verified
//
#include <hip/hip_runtime.h>
#include <math.h>

#define BB 64
#define PP 16384
#define MM 32
#define THRESH   0.35f
#define VAR0     0.1f
#define VAR1     0.2f

typedef float v2f __attribute__((ext_vector_type(2)));
typedef float v8f __attribute__((ext_vector_type(8)));

// ---------------- CDNA5 async global->LDS helpers ----------------
#if defined(__gfx1250__) && __has_builtin(__builtin_amdgcn_global_load_async_to_lds_b128)
#define HAVE_ASYNC_BUILTIN 1
#else
#define HAVE_ASYNC_BUILTIN 0
#endif

typedef int v4i __attribute__((vector_size(16)));
typedef __attribute__((address_space(1))) v4i* g4p;   // global (AS1) vec4-int ptr
typedef __attribute__((address_space(3))) v4i* l4p;   // LDS    (AS3) vec4-int ptr

__device__ __forceinline__ void async_copy_b128(const float* g, float* l) {
#if HAVE_ASYNC_BUILTIN
  __builtin_amdgcn_global_load_async_to_lds_b128((g4p)(size_t)g, (l4p)l, 0, 0);
#else
  unsigned lds = (unsigned)(size_t)(__attribute__((address_space(3))) void*)l;
  unsigned long long ga = (unsigned long long)(size_t)g;
  asm volatile("global_load_async_to_lds_b128 %0, %1, off"
               :: "v"(lds), "v"(ga) : "memory");
#endif
}

__device__ __forceinline__ void wait_async0() {
#if __has_builtin(__builtin_amdgcn_s_wait_asynccnt)
  __builtin_amdgcn_s_wait_asynccnt(0);
#else
  asm volatile("s_wait_asynccnt 0" ::: "memory");
#endif
}

// ---------------- block reductions (1024 threads, deterministic) ----------------
// Float path: tree 1024 -> 64, then one V_WMMA_F32_16X16X4_F32 with A=ones
// collapses 64 -> 16 column sums (D[0][j] = sum_k B[k][j]); thread 0 adds 16.
__device__ __forceinline__ float blockReduceF(float v, float* buf, int tid) {
  buf[tid] = v; __syncthreads();
  for (int s = 512; s >= 64; s >>= 1) {
    if (tid < s) buf[tid] += buf[tid + s];
    __syncthreads();
  }
  if (tid < 32) {                        // one full wave: EXEC all ones for WMMA
    v2f a; a.x = 1.0f; a.y = 1.0f;       // A = ones(16x4)
    v2f bv; bv.x = buf[tid]; bv.y = buf[tid + 32];  // B holds the 64 partials
    v8f c = {};
    c = __builtin_amdgcn_wmma_f32_16x16x4_f32(false, a, false, bv,
                                              (short)0, c, false, false);
    buf[64 + tid] = c[0];                // lanes 0..15: D row 0 = column sums
  }
  __syncthreads();
  if (tid == 0) {
    float s = 0.f;
    for (int i = 0; i < 16; ++i) s += buf[64 + i];
    buf[0] = s;
  }
  __syncthreads();
  float r = buf[0]; __syncthreads();
  return r;
}

__device__ __forceinline__ int blockReduceI(int v, int* buf, int tid) {
  buf[tid] = v; __syncthreads();
  for (int s = 512; s > 0; s >>= 1) {
    if (tid < s) buf[tid] += buf[tid + s];
    __syncthreads();
  }
  int r = buf[0]; __syncthreads();
  return r;
}

// ---------------- Kernel A: per-truth argmax over priors ----------------
__global__ __launch_bounds__(256) void kBestPrior(
    const float* __restrict__ targets, const float* __restrict__ priors,
    int* __restrict__ bp)
{
  const int b = blockIdx.x;
  const int tid = threadIdx.x;
  __shared__ float tb[MM][4];
  __shared__ float ta[MM];
  __shared__ float rv[256];
  __shared__ int   ri[256];

  if (tid < MM) {
    const float* t = targets + ((size_t)b * MM + tid) * 16;
    float x1 = t[0], y1 = t[1], x2 = t[2], y2 = t[3];
    tb[tid][0] = x1; tb[tid][1] = y1; tb[tid][2] = x2; tb[tid][3] = y2;
    ta[tid] = (x2 - x1) * (y2 - y1);
  }
  __syncthreads();

  const int m = tid & (MM - 1);
  const int g = tid >> 5;                 // 8 prior-groups per truth
  const float tx1 = tb[m][0], ty1 = tb[m][1], tx2 = tb[m][2], ty2 = tb[m][3];
  const float areaa = ta[m];

  float bv = -1.0f; int bi = 0;
  const float4* pr4 = (const float4*)priors;
  for (int p = g; p < PP; p += 8) {
    float4 pr = pr4[p];
    float hw = pr.z * 0.5f, hh = pr.w * 0.5f;
    float px1 = pr.x - hw, py1 = pr.y - hh, px2 = pr.x + hw, py2 = pr.y + hh;
    float areab = (px2 - px1) * (py2 - py1);
    float cw = fminf(tx2, px2) - fmaxf(tx1, px1); cw = fmaxf(cw, 0.f);
    float ch = fminf(ty2, py2) - fmaxf(ty1, py1); ch = fmaxf(ch, 0.f);
    float inter = cw * ch;
    float iou = inter / (areaa + areab - inter);
    if (iou > bv) { bv = iou; bi = p; }   // strict > -> first occurrence
  }
  rv[tid] = bv; ri[tid] = bi;
  __syncthreads();

  if (tid < MM) {
    float best = rv[tid]; int besti = ri[tid];
    for (int gg = 1; gg < 8; ++gg) {
      float v = rv[tid + (gg << 5)];
      int   ii = ri[tid + (gg << 5)];
      if (v > best || (v == best && ii < besti)) { best = v; besti = ii; }
    }
    bp[b * MM + tid] = besti;
  }
}

// ---------------- Kernel B: match + encode + BCE + top-k select, one row/WG ----------------
__global__ __launch_bounds__(1024) void kMain(
    const float* __restrict__ loc, const float* __restrict__ conf,
    const float* __restrict__ targets, const float* __restrict__ priors,
    const int* __restrict__ bp, float* __restrict__ rows)
{
  const int b = blockIdx.x;
  const int tid = threadIdx.x;

  __shared__ float    s_lc[PP];          // loss_c_all row       (64 KB)
  __shared__ float    s_conf[PP];        // async-staged logits  (64 KB)
  __shared__ float    s_t[MM * 16];      // truths + keypoints
  __shared__ float    s_area[MM];
  __shared__ unsigned s_force[PP / 32];  // force-to-2.0 bitmask
  __shared__ unsigned s_hist[256];
  __shared__ float    s_buf[1024];
  __shared__ int      s_bufi[1024];
  __shared__ unsigned s_prefix, s_kk;

  // issue async copy of the conf row into LDS (4 x b128 per thread)
  const float* crow = conf + (size_t)b * PP;
  #pragma unroll
  for (int j = 0; j < 4; ++j) {
    int idx = tid * 4 + j * 4096;
    async_copy_b128(crow + idx, &s_conf[idx]);
  }

  if (tid < PP / 32) s_force[tid] = 0u;
  if (tid < MM) {
    const float* t = targets + ((size_t)b * MM + tid) * 16;
    #pragma unroll
    for (int c = 0; c < 16; ++c) s_t[tid * 16 + c] = t[c];
    s_area[tid] = (t[2] - t[0]) * (t[3] - t[1]);
  }
  __syncthreads();
  if (tid < MM) {
    int p = bp[b * MM + tid];
    atomicOr(&s_force[p >> 5], 1u << (p & 31));
  }
  wait_async0();
  __syncthreads();

  // ---------- phase 1: per-prior match, encode, bce ----------
  float t_lossl = 0.f, t_posbce = 0.f; int t_npos = 0;
  const float4* pr4  = (const float4*)priors;
  const float4* loc4 = (const float4*)(loc + (size_t)b * PP * 16);

  for (int j = 0; j < PP / 1024; ++j) {
    int p = tid + j * 1024;
    float4 pr = pr4[p];
    float hw = pr.z * 0.5f, hh = pr.w * 0.5f;
    float px1 = pr.x - hw, py1 = pr.y - hh, px2 = pr.x + hw, py2 = pr.y + hh;
    float areab = (px2 - px1) * (py2 - py1);

    float bo = -1.f; int bi = 0;
    #pragma unroll 4
    for (int mm = 0; mm < MM; ++mm) {
      const float* t = &s_t[mm * 16];
      float cw = fminf(t[2], px2) - fmaxf(t[0], px1); cw = fmaxf(cw, 0.f);
      float ch = fminf(t[3], py2) - fmaxf(t[1], py1); ch = fmaxf(ch, 0.f);
      float inter = cw * ch;
      float iou = inter / (s_area[mm] + areab - inter);
      if (iou > bo) { bo = iou; bi = mm; }
    }
    if (s_force[p >> 5] & (1u << (p & 31))) bo = 2.0f;

    bool pos = !(bo < THRESH);
    float x = s_conf[p];
    float bce = fmaxf(x, 0.f) - (pos ? x : 0.f) + log1pf(expf(-fabsf(x)));
    s_lc[p] = pos ? 0.f : bce;

    if (pos) {
      ++t_npos; t_posbce += bce;
      const float* t = &s_t[bi * 16];
      float vx = VAR0 * pr.z, vy = VAR0 * pr.w;
      float lt[16];
      lt[0] = ((t[0] + t[2]) * 0.5f - pr.x) / vx;
      lt[1] = ((t[1] + t[3]) * 0.5f - pr.y) / vy;
      lt[2] = logf((t[2] - t[0]) / pr.z) / VAR1;
      lt[3] = logf((t[3] - t[1]) / pr.w) / VAR1;
      #pragma unroll
      for (int q = 0; q < 6; ++q) {
        lt[4 + 2 * q] = (t[4 + 2 * q] - pr.x) / vx;
        lt[5 + 2 * q] = (t[5 + 2 * q] - pr.y) / vy;
      }
      #pragma unroll
      for (int q = 0; q < 4; ++q) {
        float4 d4 = loc4[p * 4 + q];
        float dd[4] = {d4.x, d4.y, d4.z, d4.w};
        #pragma unroll
        for (int c = 0; c < 4; ++c) {
          float df = dd[c] - lt[q * 4 + c];
          float ad = fabsf(df);
          t_lossl += (ad < 1.f) ? 0.5f * df * df : ad - 0.5f;
        }
      }
    }
  }

  float sum_lossl  = blockReduceF(t_lossl, s_buf, tid);
  float sum_posbce = blockReduceF(t_posbce, s_buf, tid);
  int   num_pos    = blockReduceI(t_npos, s_bufi, tid);

  // ---------- phase 2: exact top-k sum via 4-pass radix select ----------
  int k = 3 * num_pos; if (k > PP - 1) k = PP - 1;
  float topk = 0.f;
  if (k > 0) {
    if (tid == 0) { s_prefix = 0u; s_kk = (unsigned)k; }
    __syncthreads();
    for (int pass = 0; pass < 4; ++pass) {
      int shift = 24 - 8 * pass;
      if (tid < 256) s_hist[tid] = 0u;
      __syncthreads();
      unsigned maskHi = (pass == 0) ? 0u : (0xFFFFFFFFu << (32 - 8 * pass));
      unsigned pfx = s_prefix;
      for (int j = 0; j < PP / 1024; ++j) {
        unsigned u = __float_as_uint(s_lc[tid + j * 1024]);
        if ((u & maskHi) == pfx) atomicAdd(&s_hist[(u >> shift) & 255u], 1u);
      }
      __syncthreads();
      if (tid == 0) {
        unsigned cum = 0, kk = s_kk; int sel = 0;
        for (int bin = 255; bin >= 0; --bin) {
          unsigned c = s_hist[bin];
          if (cum + c >= kk) { sel = bin; break; }
          cum += c;
        }
        s_kk = kk - cum;                       // remaining copies of pivot to take
        s_prefix = pfx | ((unsigned)sel << shift);
      }
      __syncthreads();
    }
    unsigned pivot = s_prefix;
    unsigned kk    = s_kk;
    float V = __uint_as_float(pivot);
    float ts = 0.f;
    for (int j = 0; j < PP / 1024; ++j) {
      float v = s_lc[tid + j * 1024];
      if (__float_as_uint(v) > pivot) ts += v;
    }
    float sgt = blockReduceF(ts, s_buf, tid);
    topk = sgt + (float)kk * V;                // ties at pivot: value-identical
  }

  if (tid == 0) {
    rows[b * 4 + 0] = sum_lossl;
    rows[b * 4 + 1] = sum_posbce + topk;
    rows[b * 4 + 2] = (float)num_pos;
  }
}

// ---------------- Kernel C: deterministic final combine ----------------
__global__ void kFinal(const float* __restrict__ rows, float* __restrict__ out) {
  if (threadIdx.x == 0 && blockIdx.x == 0) {
    float ll = 0.f, lc = 0.f, np = 0.f;
    for (int b = 0; b < BB; ++b) {
      ll += rows[b * 4 + 0];
      lc += rows[b * 4 + 1];
      np += rows[b * 4 + 2];
    }
    out[0] = ll / np;
    out[1] = lc / np;
  }
}

extern "C" void kernel_launch(void* const* d_in, const int* in_sizes, int n_in,
                              void* d_out, int out_size, void* d_ws, size_t ws_size,
                              hipStream_t stream) {
  (void)in_sizes; (void)n_in; (void)out_size; (void)ws_size;
  const float* loc     = (const float*)d_in[0];
  const float* conf    = (const float*)d_in[1];
  const float* targets = (const float*)d_in[2];
  const float* priors  = (const float*)d_in[3];

  int*   bp   = (int*)d_ws;
  float* rows = (float*)((char*)d_ws + BB * MM * sizeof(int));
  float* out  = (float*)d_out;

  kBestPrior<<<BB, 256, 0, stream>>>(targets, priors, bp);
  kMain<<<BB, 1024, 0, stream>>>(loc, conf, targets, priors, bp, rows);
  kFinal<<<1, 64, 0, stream>>>(rows, out);
}